// MultiStageNet_24902220382903
// MI455X (gfx1250) — compile-verified
//
#include <hip/hip_runtime.h>
#include <hip/hip_bf16.h>

typedef float v2f __attribute__((ext_vector_type(2)));
typedef float v8f __attribute__((ext_vector_type(8)));

#define BATCH     32768
#define KD        784          // 28*28
#define N_EXPERTS 5
#define ROWS_PER_WAVE 16
#define WAVES_PER_BLOCK 8
#define THREADS (WAVES_PER_BLOCK * 32)

// One wave computes a 16x16 f32 tile: [16 samples x 784] * [784 x 16 (5 real experts)]
// via V_WMMA_F32_16X16X4_F32, then selects column cat[row] per sample.
__global__ __launch_bounds__(THREADS)
void moe_wmma_kernel(const float* __restrict__ x,
                     const int*   __restrict__ cat,
                     const float* __restrict__ conv_w,
                     const float* __restrict__ conv_b,
                     const float* __restrict__ fc1_w,
                     const float* __restrict__ fc1_b,
                     float* __restrict__ out)
{
    __shared__ float tile[WAVES_PER_BLOCK][ROWS_PER_WAVE * 16];

    const int lane  = threadIdx.x & 31;
    const int wave  = threadIdx.x >> 5;
    const int gwave = blockIdx.x * WAVES_PER_BLOCK + wave;
    const int r0    = gwave * ROWS_PER_WAVE;          // first sample row of this tile

    // WMMA f32 16x16x4 fragment indexing (wave32):
    //   A: lane l -> M = l%16, K = 2*(l/16) + vgpr   (vgpr in {0,1})
    //   B: lane l -> N = l%16, K = 2*(l/16) + vgpr   (mirrors A, row striped across lanes)
    const int m  = lane & 15;          // A row within tile / B column (expert id)
    const int hk = (lane >> 4) * 2;    // K sub-offset: 0 or 2

    const bool bvalid = (m < N_EXPERTS);
    const int  nclamp = bvalid ? m : (N_EXPERTS - 1);   // keep B reads in-bounds

    // 8-byte aligned: row stride 784 floats = 3136 B, hk offset 0/8 B, k step 16 B.
    const v2f* __restrict__ ap = (const v2f*)(x      + (size_t)(r0 + m) * KD + hk);
    const v2f* __restrict__ bp = (const v2f*)(conv_w + (size_t)nclamp   * KD + hk);

    v8f acc = {};
    #pragma unroll 4
    for (int kk = 0; kk < KD / 4; ++kk) {
        v2f a = ap[2 * kk];            // x[r0+m][4kk + hk .. +1]
        v2f b = bp[2 * kk];            // conv_w[n][4kk + hk .. +1]
        b.x = bvalid ? b.x : 0.0f;     // zero-pad experts 5..15
        b.y = bvalid ? b.y : 0.0f;
        // D = A*B + C  (C = acc), f32 16x16x4
        acc = __builtin_amdgcn_wmma_f32_16x16x4_f32(
            /*neg_a=*/false, a, /*neg_b=*/false, b,
            /*c_mod=*/(short)0, acc, /*reuse_a=*/false, /*reuse_b=*/false);
    }

    // C/D layout: VGPR r, lanes 0-15 -> M=r, N=lane; lanes 16-31 -> M=r+8, N=lane-16.
    // Scatter the 16x16 tile to LDS so each sample-lane can pick its expert column.
    {
        float* t = tile[wave];
        const int mh = (lane >> 4) * 8;    // row offset for upper lane half
        const int nn = lane & 15;
        #pragma unroll
        for (int r = 0; r < 8; ++r)
            t[(mh + r) * 16 + nn] = acc[r];
    }
    __syncthreads();

    if (lane < ROWS_PER_WAVE) {
        const int   row = r0 + lane;
        const int   ci  = cat[row];
        const float s   = tile[wave][lane * 16 + ci] + conv_b[ci];
        const float o   = s * fc1_w[ci] + fc1_b[0];
        out[row] = o > 0.0f ? o : 0.0f;
    }
}

extern "C" void kernel_launch(void* const* d_in, const int* in_sizes, int n_in,
                              void* d_out, int out_size, void* d_ws, size_t ws_size,
                              hipStream_t stream) {
    const float* x    = (const float*)d_in[0];
    const int*   cat  = (const int*)  d_in[1];
    const float* cw   = (const float*)d_in[2];
    const float* cb   = (const float*)d_in[3];
    const float* f1w  = (const float*)d_in[4];
    const float* f1b  = (const float*)d_in[5];
    float*       out  = (float*)d_out;

    const int nblocks = BATCH / (ROWS_PER_WAVE * WAVES_PER_BLOCK);   // 256
    hipLaunchKernelGGL(moe_wmma_kernel, dim3(nblocks), dim3(THREADS), 0, stream,
                       x, cat, cw, cb, f1w, f1b, out);
}